// GCNModel_72619307041223
// MI455X (gfx1250) — compile-verified
//
#include <hip/hip_runtime.h>

#define IN_F  256
#define OUT_F 64

typedef __attribute__((ext_vector_type(2))) float v2f;
typedef __attribute__((ext_vector_type(8))) float v8f;

// ---------------------------------------------------------------------------
// Degree accumulation: deg_out[src[e]] += 1, deg_in[dst[e]] += 1
// ---------------------------------------------------------------------------
__global__ void gcn_degree_kernel(const int* __restrict__ src,
                                  const int* __restrict__ dst,
                                  float* __restrict__ deg_out,
                                  float* __restrict__ deg_in,
                                  int nE) {
    int e = blockIdx.x * blockDim.x + threadIdx.x;
    if (e < nE) {
        atomicAdd(&deg_out[src[e]], 1.0f);
        atomicAdd(&deg_in[dst[e]], 1.0f);
    }
}

// ---------------------------------------------------------------------------
// h = (x * deg_out^-1/2) @ W  via V_WMMA_F32_16X16X4_F32 (exact f32 path).
// Block = 256 threads = 8 waves; each wave computes a 16-row x 64-col strip.
// W is staged in LDS in a K-paired layout sW[k/2][col][2] so each B operand
// (W[kb][col], W[kb+1][col]) is one aligned 8-byte LDS load -> one VGPR pair.
// ---------------------------------------------------------------------------
__global__ __launch_bounds__(256) void gcn_gemm_wmma_kernel(
        const float* __restrict__ x,
        const float* __restrict__ W,
        const float* __restrict__ deg_out,
        float* __restrict__ h,
        int nRowTiles) {
    // sW[pair][col][parity]: pair = k>>1 (128 pairs), 64 cols, 2 floats = 64 KB
    __shared__ float sW[(IN_F / 2) * OUT_F * 2];
    for (int i = threadIdx.x; i < IN_F * OUT_F; i += 256) {
        int k   = i >> 6;        // row of W (K index)
        int col = i & 63;
        sW[(k >> 1) * (OUT_F * 2) + (col << 1) + (k & 1)] = W[i];
    }
    __syncthreads();

    const int wave = threadIdx.x >> 5;
    const int lane = threadIdx.x & 31;
    const int rt   = blockIdx.x * 8 + wave;
    if (rt >= nRowTiles) return;        // wave-uniform: EXEC stays all-ones

    const int half = lane >> 4;         // 0: lanes 0-15, 1: lanes 16-31
    const int l16  = lane & 15;
    const int row  = rt * 16 + l16;     // A-matrix: M = lane%16

    const float scale = rsqrtf(fmaxf(deg_out[row], 1.0f));
    // A layout (32-bit, 16x4): lanes 0-15 hold K=k0,k0+1; lanes 16-31 K=k0+2,k0+3
    const float* xrow = x + (size_t)row * IN_F + 2 * half;

    // Per-lane base into the K-paired W: column offset + half-wave pair offset.
    const float* sbase = sW + (l16 << 1) + half * (OUT_F * 2);

    v8f acc0 = {}, acc1 = {}, acc2 = {}, acc3 = {};

    for (int k0 = 0; k0 < IN_F; k0 += 4) {
        v2f a;
        a.x = xrow[k0]     * scale;
        a.y = xrow[k0 + 1] * scale;

        // pair index for this k-step: (k0>>1) + half, stride OUT_F*2 floats
        const float* p = sbase + (k0 >> 1) * (OUT_F * 2);
        v2f b0 = *(const v2f*)(p);        // cols  0..15
        v2f b1 = *(const v2f*)(p + 32);   // cols 16..31
        v2f b2 = *(const v2f*)(p + 64);   // cols 32..47
        v2f b3 = *(const v2f*)(p + 96);   // cols 48..63

        acc0 = __builtin_amdgcn_wmma_f32_16x16x4_f32(false, a, false, b0, (short)0, acc0, false, false);
        acc1 = __builtin_amdgcn_wmma_f32_16x16x4_f32(false, a, false, b1, (short)0, acc1, false, false);
        acc2 = __builtin_amdgcn_wmma_f32_16x16x4_f32(false, a, false, b2, (short)0, acc2, false, false);
        acc3 = __builtin_amdgcn_wmma_f32_16x16x4_f32(false, a, false, b3, (short)0, acc3, false, false);
    }

    // C/D 16x16 layout: VGPR r -> (M=r, N=lane) for lanes 0-15, (M=r+8) for 16-31.
    const int outRowBase = rt * 16 + 8 * half;
    for (int r = 0; r < 8; ++r) {
        float* hr = h + (size_t)(outRowBase + r) * OUT_F + l16;
        hr[0]  = acc0[r];
        hr[16] = acc1[r];
        hr[32] = acc2[r];
        hr[48] = acc3[r];
    }
}

// ---------------------------------------------------------------------------
// SpMM: out[dst[e]] += h[src[e]]  (16 threads per edge, float4 per thread).
// h (25.6 MB) and out (25.6 MB) are L2-resident (192 MB L2).
// ---------------------------------------------------------------------------
__global__ void gcn_spmm_kernel(const int* __restrict__ src,
                                const int* __restrict__ dst,
                                const float* __restrict__ h,
                                float* __restrict__ out,
                                int nE) {
    int t  = blockIdx.x * blockDim.x + threadIdx.x;
    int e  = t >> 4;
    int c4 = (t & 15) * 4;
    if (e >= nE) return;
    int s = src[e];
    int d = dst[e];
    const float4 v = *(const float4*)(h + (size_t)s * OUT_F + c4);
    float* o = out + (size_t)d * OUT_F + c4;
    atomicAdd(o + 0, v.x);
    atomicAdd(o + 1, v.y);
    atomicAdd(o + 2, v.z);
    atomicAdd(o + 3, v.w);
}

// ---------------------------------------------------------------------------
// out = out * deg_in^-1/2 + b  (in place)
// ---------------------------------------------------------------------------
__global__ void gcn_finalize_kernel(float* __restrict__ out,
                                    const float* __restrict__ deg_in,
                                    const float* __restrict__ bias,
                                    int total) {
    int i = blockIdx.x * blockDim.x + threadIdx.x;
    if (i >= total) return;
    int row = i >> 6;       // OUT_F == 64
    int col = i & 63;
    out[i] = out[i] * rsqrtf(fmaxf(deg_in[row], 1.0f)) + bias[col];
}

// ---------------------------------------------------------------------------
// Launch
// ---------------------------------------------------------------------------
extern "C" void kernel_launch(void* const* d_in, const int* in_sizes, int n_in,
                              void* d_out, int out_size, void* d_ws, size_t ws_size,
                              hipStream_t stream) {
    const float* x    = (const float*)d_in[0];
    const int*   src  = (const int*)d_in[1];
    const int*   dst  = (const int*)d_in[2];
    const float* W    = (const float*)d_in[3];
    const float* bias = (const float*)d_in[4];

    const int nNodes = in_sizes[0] / IN_F;   // 100000
    const int nE     = in_sizes[1];          // 1000000

    // Workspace layout: deg_out[N] | deg_in[N] | h[N*OUT_F]
    float* deg_out = (float*)d_ws;
    float* deg_in  = deg_out + nNodes;
    float* h       = deg_in + nNodes;
    float* out     = (float*)d_out;

    // Zero accumulators (memset nodes are graph-capture safe).
    hipMemsetAsync(d_ws, 0, (size_t)2 * nNodes * sizeof(float), stream);
    hipMemsetAsync(d_out, 0, (size_t)out_size * sizeof(float), stream);

    gcn_degree_kernel<<<(nE + 255) / 256, 256, 0, stream>>>(src, dst, deg_out, deg_in, nE);

    const int nRowTiles = nNodes / 16;       // 100000 % 16 == 0
    gcn_gemm_wmma_kernel<<<(nRowTiles + 7) / 8, 256, 0, stream>>>(x, W, deg_out, h, nRowTiles);

    const long long spThreads = (long long)nE * 16;
    gcn_spmm_kernel<<<(unsigned)((spThreads + 255) / 256), 256, 0, stream>>>(src, dst, h, out, nE);

    const int total = nNodes * OUT_F;
    gcn_finalize_kernel<<<(total + 255) / 256, 256, 0, stream>>>(out, deg_in, bias, total);
}